// LayerRegression_43860206027513
// MI455X (gfx1250) — compile-verified
//
#include <hip/hip_runtime.h>
#include <hip/hip_bf16.h>

#define N_OPS 3000
#define NCHUNK 32

typedef __attribute__((ext_vector_type(2))) float v2f;
typedef __attribute__((ext_vector_type(8))) float v8f;

// ---------------------------------------------------------------------------
// Phase 1: per-chunk partial unitary, one wave32 per chunk, lane = row,
// 32 complex amplitudes (64 floats) per lane held in VGPRs.
// ---------------------------------------------------------------------------

template <int MASK>
__device__ __forceinline__ void apply1q(float* re, float* im,
                                        float u00r, float u00i, float u01r, float u01i,
                                        float u10r, float u10i, float u11r, float u11i) {
#pragma unroll
  for (int n = 0; n < 32; ++n) {
    if (n & MASK) continue;           // compile-time filtered after unroll
    const int m = n | MASK;
    float ar = re[n], ai = im[n], br = re[m], bi = im[m];
    re[n] = u00r * ar - u00i * ai + u01r * br - u01i * bi;
    im[n] = u00r * ai + u00i * ar + u01r * bi + u01i * br;
    re[m] = u10r * ar - u10i * ai + u11r * br - u11i * bi;
    im[m] = u10r * ai + u10i * ar + u11r * bi + u11i * br;
  }
}

template <int CMASK, int TMASK>
__device__ __forceinline__ void applyCnot(float* re, float* im) {
#pragma unroll
  for (int n = 0; n < 32; ++n) {
    if (!(n & CMASK) || (n & TMASK)) continue;
    const int m = n | TMASK;
    float tr = re[n], ti = im[n];
    re[n] = re[m]; im[n] = im[m];
    re[m] = tr;    im[m] = ti;
  }
}

__global__ void __launch_bounds__(32)
chunk_kernel(const float* __restrict__ angles, const int* __restrict__ gtypes,
             const int* __restrict__ w0s, const int* __restrict__ w1s,
             float* __restrict__ out) {
  const int k = blockIdx.x;
  const int lane = threadIdx.x;   // row index of the partial unitary

  float re[32], im[32];
#pragma unroll
  for (int n = 0; n < 32; ++n) { re[n] = (n == lane) ? 1.0f : 0.0f; im[n] = 0.0f; }

  const int start = (k * N_OPS) / NCHUNK;
  const int end   = ((k + 1) * N_OPS) / NCHUNK;

  for (int i = start; i < end; ++i) {
    const int g  = gtypes[i];
    const int w0 = w0s[i];
    if (g == 3) {
      const int w1 = w1s[i];
#define CN(c, t) case ((c)*5 + (t)): applyCnot<(1 << (4 - (c))), (1 << (4 - (t)))>(re, im); break;
      switch (w0 * 5 + w1) {
        CN(0,1) CN(0,2) CN(0,3) CN(0,4)
        CN(1,0) CN(1,2) CN(1,3) CN(1,4)
        CN(2,0) CN(2,1) CN(2,3) CN(2,4)
        CN(3,0) CN(3,1) CN(3,2) CN(3,4)
        CN(4,0) CN(4,1) CN(4,2) CN(4,3)
        default: break;
      }
#undef CN
    } else {
      float s, c;
      sincosf(0.5f * angles[i], &s, &c);
      float u00r, u00i, u01r, u01i, u10r, u10i, u11r, u11i;
      if (g == 0) {        // RX: [[c, -i s],[-i s, c]]
        u00r = c; u00i = 0.f; u01r = 0.f; u01i = -s;
        u10r = 0.f; u10i = -s; u11r = c; u11i = 0.f;
      } else if (g == 1) { // RY: [[c, -s],[s, c]]
        u00r = c; u00i = 0.f; u01r = -s; u01i = 0.f;
        u10r = s; u10i = 0.f; u11r = c; u11i = 0.f;
      } else {             // RZ: diag(c - i s, c + i s)
        u00r = c; u00i = -s; u01r = 0.f; u01i = 0.f;
        u10r = 0.f; u10i = 0.f; u11r = c; u11i = s;
      }
      switch (w0) {        // wire w -> column bit (4-w)
        case 0: apply1q<16>(re, im, u00r,u00i,u01r,u01i,u10r,u10i,u11r,u11i); break;
        case 1: apply1q< 8>(re, im, u00r,u00i,u01r,u01i,u10r,u10i,u11r,u11i); break;
        case 2: apply1q< 4>(re, im, u00r,u00i,u01r,u01i,u10r,u10i,u11r,u11i); break;
        case 3: apply1q< 2>(re, im, u00r,u00i,u01r,u01i,u10r,u10i,u11r,u11i); break;
        default: apply1q<1>(re, im, u00r,u00i,u01r,u01i,u10r,u10i,u11r,u11i); break;
      }
    }
  }

  // slot layout: slot k = 2048 floats: re plane [32*32] then im plane [32*32]
  float* pr = out + (size_t)k * 2048;
  float* pi = pr + 1024;
#pragma unroll
  for (int n = 0; n < 32; ++n) {
    pr[lane * 32 + n] = re[n];
    pi[lane * 32 + n] = im[n];
  }
}

// ---------------------------------------------------------------------------
// Phase 2: pairwise complex 32x32 matmul C = A*B via V_WMMA_F32_16X16X4_F32.
// Block = 128 threads = 4 waves; wave t owns 16x16 output tile (t>>1, t&1).
// Cr = Ar*Br + (-Ai)*Bi ; Ci = Ar*Bi + Ai*Br  (f32 WMMA has no A/B negate,
// so Ai is negated with VALU for the Cr term).
// ---------------------------------------------------------------------------
__global__ void __launch_bounds__(128)
combine_kernel(const float* __restrict__ in, float* __restrict__ out, int finalOut) {
  const int p    = blockIdx.x;
  const int wave = threadIdx.x >> 5;
  const int lane = threadIdx.x & 31;
  const int row   = lane & 15;   // M (A/C) or N (B) within tile
  const int khalf = lane >> 4;   // lane-group select

  const float* Ar = in + (size_t)(2 * p) * 2048;
  const float* Ai = Ar + 1024;
  const float* Br = in + (size_t)(2 * p + 1) * 2048;
  const float* Bi = Br + 1024;

  const int ti = (wave >> 1) * 16;  // C row tile base
  const int tj = (wave & 1) * 16;   // C col tile base

  v8f cr = {0.f, 0.f, 0.f, 0.f, 0.f, 0.f, 0.f, 0.f};
  v8f ci = {0.f, 0.f, 0.f, 0.f, 0.f, 0.f, 0.f, 0.f};

#pragma unroll
  for (int kt = 0; kt < 8; ++kt) {
    const int kbase = kt * 4;
    v2f ar, ai, br, bi;
#pragma unroll
    for (int v = 0; v < 2; ++v) {
      // A 16x4 fragment: lane L, vgpr v -> M = L%16, K = 2*(L/16)+v
      const int ak = kbase + 2 * khalf + v;
      ar[v] = Ar[(ti + row) * 32 + ak];
      ai[v] = Ai[(ti + row) * 32 + ak];
      // B 4x16 fragment: vgpr v, lanes0-15: K=v, N=lane; lanes16-31: K=v+2
      const int bk = kbase + 2 * khalf + v;
      br[v] = Br[bk * 32 + (tj + row)];
      bi[v] = Bi[bk * 32 + (tj + row)];
    }
    v2f nai = -ai;
    cr = __builtin_amdgcn_wmma_f32_16x16x4_f32(false, ar,  false, br, (short)0, cr, false, false);
    cr = __builtin_amdgcn_wmma_f32_16x16x4_f32(false, nai, false, bi, (short)0, cr, false, false);
    ci = __builtin_amdgcn_wmma_f32_16x16x4_f32(false, ar,  false, bi, (short)0, ci, false, false);
    ci = __builtin_amdgcn_wmma_f32_16x16x4_f32(false, ai,  false, br, (short)0, ci, false, false);
  }

  // C/D layout: vgpr v, lanes0-15: M=v, N=lane; lanes16-31: M=v+8, N=lane-16
#pragma unroll
  for (int v = 0; v < 8; ++v) {
    const int m = ti + v + 8 * khalf;
    const int n = tj + row;
    if (finalOut) { // interleaved complex64 row-major (re, im)
      out[(size_t)(m * 32 + n) * 2 + 0] = cr[v];
      out[(size_t)(m * 32 + n) * 2 + 1] = ci[v];
    } else {
      out[(size_t)p * 2048 + m * 32 + n]        = cr[v];
      out[(size_t)p * 2048 + 1024 + m * 32 + n] = ci[v];
    }
  }
}

// ---------------------------------------------------------------------------
extern "C" void kernel_launch(void* const* d_in, const int* in_sizes, int n_in,
                              void* d_out, int out_size, void* d_ws, size_t ws_size,
                              hipStream_t stream) {
  (void)in_sizes; (void)n_in; (void)out_size; (void)ws_size;
  // setup_inputs order: x, angles, gtypes, w0s, w1s
  const float* angles = (const float*)d_in[1];
  const int*   gtypes = (const int*)d_in[2];
  const int*   w0s    = (const int*)d_in[3];
  const int*   w1s    = (const int*)d_in[4];

  float* bufA = (float*)d_ws;             // 32 slots * 2048 floats = 256 KB
  float* bufB = bufA + 32 * 2048;         // 16 slots               = 128 KB
  float* outF = (float*)d_out;            // 2048 floats (complex64 32x32)

  chunk_kernel<<<NCHUNK, 32, 0, stream>>>(angles, gtypes, w0s, w1s, bufA);
  combine_kernel<<<16, 128, 0, stream>>>(bufA, bufB, 0);  // 32 -> 16
  combine_kernel<<< 8, 128, 0, stream>>>(bufB, bufA, 0);  // 16 ->  8
  combine_kernel<<< 4, 128, 0, stream>>>(bufA, bufB, 0);  //  8 ->  4
  combine_kernel<<< 2, 128, 0, stream>>>(bufB, bufA, 0);  //  4 ->  2
  combine_kernel<<< 1, 128, 0, stream>>>(bufA, outF, 1);  //  2 ->  1 (final)
}